// ConceptGAT_71451075936978
// MI455X (gfx1250) — compile-verified
//
#include <hip/hip_runtime.h>
#include <math.h>

// ---------------------------------------------------------------------------
// ConceptGAT forward for MI455X (gfx1250, wave32).
//  - GEMMs run on the matrix unit via V_WMMA_F32_16X16X4_F32 (full fp32).
//  - Edge softmax/aggregation uses L2-resident gathers + fp32 global atomics.
// ---------------------------------------------------------------------------

typedef __attribute__((ext_vector_type(2))) float v2f;
typedef __attribute__((ext_vector_type(8))) float v8f;

__device__ __forceinline__ float atomic_add_f32(float* p, float v) {
  return __hip_atomic_fetch_add(p, v, __ATOMIC_RELAXED, __HIP_MEMORY_SCOPE_AGENT);
}

// Float atomic max via integer ordering trick (valid for all finite floats &
// +/-inf): non-negative -> signed int max; negative -> unsigned int min.
__device__ __forceinline__ void atomic_max_f32(float* p, float v) {
  if (v >= 0.0f) atomicMax((int*)p, __float_as_int(v));
  else           atomicMin((unsigned int*)p, __float_as_uint(v));
}

// ---------------------------------------------------------------------------
// fp32 GEMM: C[M,N] = A[M,K] * B[K,N], one wave per 16x64 output strip.
// A-fragment (16x4): lane<16 -> A[M=lane][k0..k0+1], lane>=16 -> A[M=lane-16][k0+2..k0+3]
// B-fragment (4x16): lane<16 -> B[k0][N=col],B[k0+1][col]; lane>=16 -> rows k0+2,k0+3
// C/D (16x16): vgpr r, lane l -> D[r + 8*(l>=16)][l&15]
// M%16==0, N%64==0, K%4==0 (holds for all shapes here).
// ---------------------------------------------------------------------------
__global__ __launch_bounds__(32) void wmma_gemm_f32(
    const float* __restrict__ A, const float* __restrict__ B,
    float* __restrict__ C, int M, int N, int K)
{
  const int lane = threadIdx.x & 31;
  const int half = lane >> 4;      // 0: K=k0,k0+1   1: K=k0+2,k0+3
  const int l15  = lane & 15;
  const int tm   = blockIdx.x;     // 16-row tile index
  const int tn   = blockIdx.y << 6; // 64-col strip base

  v8f c0 = {}; v8f c1 = {}; v8f c2 = {}; v8f c3 = {};

  const float* Arow = A + (size_t)(tm * 16 + l15) * K;
  for (int k0 = 0; k0 < K; k0 += 4) {
    const int ka = k0 + (half << 1);
    v2f a;
    a.x = Arow[ka];
    a.y = Arow[ka + 1];
    const float* Bp = B + (size_t)ka * N + tn + l15;
    v2f b0, b1, b2, b3;
    b0.x = Bp[0];   b0.y = Bp[N];
    b1.x = Bp[16];  b1.y = Bp[N + 16];
    b2.x = Bp[32];  b2.y = Bp[N + 32];
    b3.x = Bp[48];  b3.y = Bp[N + 48];
    c0 = __builtin_amdgcn_wmma_f32_16x16x4_f32(false, a, false, b0, (short)0, c0, false, false);
    c1 = __builtin_amdgcn_wmma_f32_16x16x4_f32(false, a, false, b1, (short)0, c1, false, false);
    c2 = __builtin_amdgcn_wmma_f32_16x16x4_f32(false, a, false, b2, (short)0, c2, false, false);
    c3 = __builtin_amdgcn_wmma_f32_16x16x4_f32(false, a, false, b3, (short)0, c3, false, false);
  }

  const int orow = tm * 16 + (half << 3);
  for (int r = 0; r < 8; ++r) {
    float* Cp = C + (size_t)(orow + r) * N + tn + l15;
    Cp[0]  = c0[r];
    Cp[16] = c1[r];
    Cp[32] = c2[r];
    Cp[48] = c3[r];
  }
}

// ---------------------------------------------------------------------------
// a_src[n,h] = sum_c xp[n,h,c]*att_src[h,c]; same for a_dst. One wave / (n,h).
// ---------------------------------------------------------------------------
__global__ __launch_bounds__(256) void node_dots(
    const float* __restrict__ xp, const float* __restrict__ att_src,
    const float* __restrict__ att_dst, float* __restrict__ asrc,
    float* __restrict__ adst, int Nn, int H, int C)
{
  const int wid  = (int)((blockIdx.x * blockDim.x + threadIdx.x) >> 5);
  const int lane = threadIdx.x & 31;
  if (wid >= Nn * H) return;
  const int n = wid / H;
  const int h = wid - n * H;

  const float* xr = xp + (size_t)n * H * C + (size_t)h * C;
  const float* as = att_src + (size_t)h * C;
  const float* ad = att_dst + (size_t)h * C;
  float s = 0.f, d = 0.f;
  for (int c = lane; c < C; c += 32) {
    float v = xr[c];
    s += v * as[c];
    d += v * ad[c];
  }
  for (int off = 16; off > 0; off >>= 1) {
    s += __shfl_down(s, off, 32);
    d += __shfl_down(d, off, 32);
  }
  if (lane == 0) { asrc[wid] = s; adst[wid] = d; }
}

__global__ void fill_f32(float* __restrict__ p, float v, size_t n) {
  size_t i = (size_t)blockIdx.x * blockDim.x + threadIdx.x;
  size_t stride = (size_t)gridDim.x * blockDim.x;
  for (; i < n; i += stride) p[i] = v;
}

// Pass 1: per-dst running max of leaky-relu'd attention logits.
__global__ __launch_bounds__(256) void edge_max(
    const int* __restrict__ src, const int* __restrict__ dst, int E0, int Nn,
    const float* __restrict__ asrc, const float* __restrict__ adst,
    float* __restrict__ amax, int H)
{
  const int e = blockIdx.x * blockDim.x + threadIdx.x;
  if (e >= E0 + Nn) return;
  const int s = (e < E0) ? src[e] : (e - E0);
  const int d = (e < E0) ? dst[e] : (e - E0);
  for (int h = 0; h < H; ++h) {
    float a = asrc[s * H + h] + adst[d * H + h];
    a = (a > 0.f) ? a : 0.2f * a;               // leaky_relu, slope 0.2
    atomic_max_f32(&amax[d * H + h], a);
  }
}

// Pass 2: ex = exp(alpha - amax[dst]); store per-edge ex, accumulate denom.
__global__ __launch_bounds__(256) void edge_exp(
    const int* __restrict__ src, const int* __restrict__ dst, int E0, int Nn,
    const float* __restrict__ asrc, const float* __restrict__ adst,
    const float* __restrict__ amax, float* __restrict__ den,
    float* __restrict__ ex, int H)
{
  const int e = blockIdx.x * blockDim.x + threadIdx.x;
  if (e >= E0 + Nn) return;
  const int s = (e < E0) ? src[e] : (e - E0);
  const int d = (e < E0) ? dst[e] : (e - E0);
  for (int h = 0; h < H; ++h) {
    float a = asrc[s * H + h] + adst[d * H + h];
    a = (a > 0.f) ? a : 0.2f * a;
    float v = __expf(a - amax[d * H + h]);
    ex[(size_t)e * H + h] = v;
    atomic_add_f32(&den[d * H + h], v);
  }
}

// Pass 3: out[dst] += xp[src] * (ex/den[dst]). One block per edge; features
// striped across 256 threads. C must be 256 (h = f>>8). xp fits in L2.
__global__ __launch_bounds__(256) void edge_aggregate(
    const int* __restrict__ src, const int* __restrict__ dst, int E0,
    const float* __restrict__ xp, const float* __restrict__ ex,
    const float* __restrict__ den, float* __restrict__ out, int H)
{
  const int e = blockIdx.x;
  const int s = (e < E0) ? src[e] : (e - E0);
  const int d = (e < E0) ? dst[e] : (e - E0);
  const int F = H << 8;                         // H * 256
  const size_t eb = (size_t)e * H;
  const size_t sb = (size_t)s * F;
  const size_t db = (size_t)d * F;
  for (int f = threadIdx.x; f < F; f += 256) {
    const int h = f >> 8;
    const float w = ex[eb + h] / den[d * H + h];
    atomic_add_f32(&out[db + f], xp[sb + f] * w);
  }
}

// y = elu(y + bias[f % F])  (alpha=1 -> expm1 on negative side)
__global__ void bias_elu(float* __restrict__ y, const float* __restrict__ b,
                         int F, size_t n) {
  size_t i = (size_t)blockIdx.x * blockDim.x + threadIdx.x;
  size_t stride = (size_t)gridDim.x * blockDim.x;
  for (; i < n; i += stride) {
    int f = (int)(i - (i / F) * F);
    float v = y[i] + b[f];
    y[i] = (v > 0.f) ? v : expm1f(v);
  }
}

// y += bias[f & (F-1)]  (F power of two)
__global__ void bias_add(float* __restrict__ y, const float* __restrict__ b,
                         int Fmask, size_t n) {
  size_t i = (size_t)blockIdx.x * blockDim.x + threadIdx.x;
  size_t stride = (size_t)gridDim.x * blockDim.x;
  for (; i < n; i += stride) y[i] += b[(int)i & Fmask];
}

// ---------------------------------------------------------------------------

extern "C" void kernel_launch(void* const* d_in, const int* in_sizes, int n_in,
                              void* d_out, int out_size, void* d_ws, size_t ws_size,
                              hipStream_t stream) {
  const float* x   = (const float*)d_in[0];
  const int*   ei  = (const int*)d_in[1];   // [2, E] int32
  const float* W1  = (const float*)d_in[2];
  const float* as1 = (const float*)d_in[3];
  const float* ad1 = (const float*)d_in[4];
  const float* b1  = (const float*)d_in[5];
  const float* W2  = (const float*)d_in[6];
  const float* as2 = (const float*)d_in[7];
  const float* ad2 = (const float*)d_in[8];
  const float* b2  = (const float*)d_in[9];

  const int K1 = 768, F1 = 768, H1 = 3;     // layer 1: heads=3, C=256
  const int F2 = 256, H2 = 1;               // layer 2: heads=1, C=256
  const int Nn = in_sizes[0] / K1;          // 20000
  const int E0 = in_sizes[1] / 2;           // 320000
  const int ET = E0 + Nn;                   // + self loops

  const int* srcI = ei;
  const int* dstI = ei + E0;

  // workspace carve-up (floats)
  float* ws = (float*)d_ws;
  float* xp1   = ws; ws += (size_t)Nn * F1;   // 61.4 MB
  float* h1    = ws; ws += (size_t)Nn * F1;   // layer-1 output / ELU / GEMM2 input
  float* xp2   = ws; ws += (size_t)Nn * F2;
  float* asrc1 = ws; ws += (size_t)Nn * H1;
  float* adst1 = ws; ws += (size_t)Nn * H1;
  float* amax1 = ws; ws += (size_t)Nn * H1;
  float* den1  = ws; ws += (size_t)Nn * H1;
  float* asrc2 = ws; ws += Nn;
  float* adst2 = ws; ws += Nn;
  float* amax2 = ws; ws += Nn;
  float* den2  = ws; ws += Nn;
  float* ex1   = ws; ws += (size_t)ET * H1;
  float* ex2   = ws; ws += (size_t)ET;

  const float NEG_INF = -__builtin_inff();
  const int FB = 512;                       // blocks for grid-stride fills

  // ---------------- Layer 1 ----------------
  {
    dim3 g(Nn / 16, F1 / 64);               // 1250 x 12 waves
    wmma_gemm_f32<<<g, 32, 0, stream>>>(x, W1, xp1, Nn, F1, K1);
  }
  node_dots<<<(Nn * H1 + 7) / 8, 256, 0, stream>>>(xp1, as1, ad1, asrc1, adst1, Nn, H1, 256);
  fill_f32<<<FB, 256, 0, stream>>>(amax1, NEG_INF, (size_t)Nn * H1);
  fill_f32<<<FB, 256, 0, stream>>>(den1, 0.f, (size_t)Nn * H1);
  fill_f32<<<FB, 256, 0, stream>>>(h1, 0.f, (size_t)Nn * F1);
  edge_max<<<(ET + 255) / 256, 256, 0, stream>>>(srcI, dstI, E0, Nn, asrc1, adst1, amax1, H1);
  edge_exp<<<(ET + 255) / 256, 256, 0, stream>>>(srcI, dstI, E0, Nn, asrc1, adst1, amax1, den1, ex1, H1);
  edge_aggregate<<<ET, 256, 0, stream>>>(srcI, dstI, E0, xp1, ex1, den1, h1, H1);
  bias_elu<<<FB, 256, 0, stream>>>(h1, b1, F1, (size_t)Nn * F1);

  // ---------------- Layer 2 ----------------
  {
    dim3 g(Nn / 16, F2 / 64);               // 1250 x 4 waves
    wmma_gemm_f32<<<g, 32, 0, stream>>>(h1, W2, xp2, Nn, F2, F1);
  }
  node_dots<<<(Nn + 7) / 8, 256, 0, stream>>>(xp2, as2, ad2, asrc2, adst2, Nn, H2, 256);
  fill_f32<<<FB, 256, 0, stream>>>(amax2, NEG_INF, (size_t)Nn);
  fill_f32<<<FB, 256, 0, stream>>>(den2, 0.f, (size_t)Nn);
  fill_f32<<<FB, 256, 0, stream>>>((float*)d_out, 0.f, (size_t)Nn * F2);
  edge_max<<<(ET + 255) / 256, 256, 0, stream>>>(srcI, dstI, E0, Nn, asrc2, adst2, amax2, H2);
  edge_exp<<<(ET + 255) / 256, 256, 0, stream>>>(srcI, dstI, E0, Nn, asrc2, adst2, amax2, den2, ex2, H2);
  edge_aggregate<<<ET, 256, 0, stream>>>(srcI, dstI, E0, xp2, ex2, den2, (float*)d_out, H2);
  bias_add<<<FB, 256, 0, stream>>>((float*)d_out, b2, F2 - 1, (size_t)Nn * F2);
}